// LGCNEncoder_86311662780537
// MI455X (gfx1250) — compile-verified
//
#include <hip/hip_runtime.h>
#include <cmath>

#define NUSERS  50000
#define NITEMS  100000
#define NNODES  (NUSERS + NITEMS)   // 150000
#define DIM     64
#define NEDGES  2400000
#define NLAYERS 3
#define ALPHAM  1.0f
#define EPSV    1e-6f

typedef float v2f __attribute__((ext_vector_type(2)));
typedef float v4f __attribute__((ext_vector_type(4)));
typedef float v8f __attribute__((ext_vector_type(8)));

#ifdef __has_builtin
#if __has_builtin(__builtin_amdgcn_wmma_f32_16x16x4_f32)
#define USE_WMMA_REDUCE 1
#endif
#endif

// ---------------------------------------------------------------------------
// K1: emb = concat(user_emb, item_emb); acc = emb (snapshot 0); embn = 0
// ---------------------------------------------------------------------------
__global__ void lgcn_init(const float* __restrict__ ue, const float* __restrict__ ie,
                          float* __restrict__ emb, float* __restrict__ embn,
                          float* __restrict__ acc) {
    int i = blockIdx.x * blockDim.x + threadIdx.x;           // float4 index
    const int total4 = NNODES * (DIM / 4);
    if (i >= total4) return;
    const int usz4 = NUSERS * (DIM / 4);
    v4f v = (i < usz4) ? ((const v4f*)ue)[i] : ((const v4f*)ie)[i - usz4];
    ((v4f*)emb)[i]  = v;
    ((v4f*)acc)[i]  = v;
    v4f z = {0.f, 0.f, 0.f, 0.f};
    ((v4f*)embn)[i] = z;
}

// ---------------------------------------------------------------------------
// K2: scatter  embn[row] += val * emb[col]   (one edge per wave32,
//     lane handles 2 dims as float2; edge metadata via scalar path)
// ---------------------------------------------------------------------------
__global__ void lgcn_scatter(const float* __restrict__ vals, const int* __restrict__ rows,
                             const int* __restrict__ cols, const float* __restrict__ emb,
                             float* __restrict__ embn) {
    int wid = (blockIdx.x << 3) + (threadIdx.x >> 5);        // 8 waves / 256-thread block
    if (wid >= NEDGES) return;
    const int lane = threadIdx.x & 31;
#ifdef __HIP_DEVICE_COMPILE__
    const int e = __builtin_amdgcn_readfirstlane(wid);       // force SGPR -> s_load metadata
#else
    const int e = wid;
#endif
    const float v = vals[e];
    const int   r = rows[e];
    const int   c = cols[e];
    v2f s = ((const v2f*)(emb))[(size_t)c * (DIM / 2) + lane];
    float* dst = embn + (size_t)r * DIM + lane * 2;
    atomicAdd(dst + 0, v * s.x);
    atomicAdd(dst + 1, v * s.y);
}

// ---------------------------------------------------------------------------
// K3: growth gating. One wave per 16 nodes.
//     Phase 1: squared norms of (old - new + eps) rows via V_WMMA_F32_16X16X4_F32
//              with a ones B-matrix (D[m][*] = row-sum), accumulated over K.
//     Phase 2: blend emb = wo*old + wn*new, acc += emb, embn = 0 (for next layer).
//     Guards are wave-uniform -> EXEC all-ones at every WMMA.
// ---------------------------------------------------------------------------
__global__ void lgcn_gate(float* __restrict__ emb, float* __restrict__ embn,
                          float* __restrict__ acc) {
    const int groups = NNODES / 16;                          // 9375
    const int group  = blockIdx.x * (blockDim.x >> 5) + (threadIdx.x >> 5);
    if (group >= groups) return;                             // wave-uniform exit
    const int lane  = threadIdx.x & 31;
    const int node0 = group * 16;
    const int m     = lane & 15;                             // node within group (A-frag row)
    const int kb    = (lane >> 4) << 1;                      // K sub-offset: 0 (lo half) / 2 (hi half)
    const int half  = lane >> 4;

    const float* oldr = emb  + (size_t)(node0 + m) * DIM;
    const float* newr = embn + (size_t)(node0 + m) * DIM;

#ifdef USE_WMMA_REDUCE
    v8f cmat = {};
    v2f ones; ones.x = 1.0f; ones.y = 1.0f;                  // B = 4x16 all-ones
    #pragma unroll
    for (int k0 = 0; k0 < DIM; k0 += 4) {
        v2f o  = *(const v2f*)(oldr + k0 + kb);
        v2f nw = *(const v2f*)(newr + k0 + kb);
        float d0 = o.x - nw.x + EPSV;
        float d1 = o.y - nw.y + EPSV;
        v2f a; a.x = d0 * d0; a.y = d1 * d1;                 // A-frag per ISA 16x4 f32 layout
        cmat = __builtin_amdgcn_wmma_f32_16x16x4_f32(
            false, a, false, ones, (short)0, cmat, false, false);
    }
    // D layout: VGPR r, lanes 0-15 -> M=r ; lanes 16-31 -> M=8+r (value uniform across N)
    const int l16 = lane & 15;
    #pragma unroll
    for (int r2 = 0; r2 < 8; ++r2) {
        const int   mm = r2 + half * 8;
        const float ns = cmat[r2];                           // node mm squared norm
        const float os = sqrtf(ns);
        const float dn = ALPHAM * log1pf(os);
        const float da = 1.0f + dn;
        const float wo = 1.0f / da;
        const float wn = dn / da;
        const size_t base = (size_t)(node0 + mm) * DIM + l16 * 4;
        v4f o  = *(const v4f*)(emb  + base);
        v4f nn = *(const v4f*)(embn + base);
        v4f outv = wo * o + wn * nn;
        *(v4f*)(emb + base) = outv;
        v4f a4 = *(const v4f*)(acc + base);
        a4 += outv;
        *(v4f*)(acc + base) = a4;
        v4f z = {0.f, 0.f, 0.f, 0.f};
        *(v4f*)(embn + base) = z;                            // pre-zero for next layer
    }
#else
    // Fallback: lanes L / L+16 split K{0,1}/{2,3} per chunk; combine with xor-16.
    float partial = 0.f;
    #pragma unroll
    for (int k0 = 0; k0 < DIM; k0 += 4) {
        v2f o  = *(const v2f*)(oldr + k0 + kb);
        v2f nw = *(const v2f*)(newr + k0 + kb);
        float d0 = o.x - nw.x + EPSV;
        float d1 = o.y - nw.y + EPSV;
        partial = fmaf(d0, d0, partial);
        partial = fmaf(d1, d1, partial);
    }
    partial += __shfl_xor(partial, 16);
    const float os = sqrtf(partial);
    const float dn = ALPHAM * log1pf(os);
    const float da = 1.0f + dn;
    const float wo = 1.0f / da;
    const float wn = dn / da;
    const size_t base = (size_t)(node0 + m) * DIM + half * 32;
    #pragma unroll
    for (int q = 0; q < 8; ++q) {
        const size_t idx = base + q * 4;
        v4f o  = *(const v4f*)(emb  + idx);
        v4f nn = *(const v4f*)(embn + idx);
        v4f outv = wo * o + wn * nn;
        *(v4f*)(emb + idx) = outv;
        v4f a4 = *(const v4f*)(acc + idx);
        a4 += outv;
        *(v4f*)(acc + idx) = a4;
        v4f z = {0.f, 0.f, 0.f, 0.f};
        *(v4f*)(embn + idx) = z;
    }
#endif
}

// ---------------------------------------------------------------------------
// K4: final = acc/4 ; gather user/item rows into d_out [2048 x 64]
// ---------------------------------------------------------------------------
__global__ void lgcn_gather(const float* __restrict__ acc, const int* __restrict__ uid,
                            const int* __restrict__ iid, float* __restrict__ out) {
    int i = blockIdx.x * blockDim.x + threadIdx.x;
    if (i >= 2048 * DIM) return;
    const int j = i >> 6;
    const int d = i & 63;
    const size_t node = (j < 1024) ? (size_t)uid[j] : (size_t)(NUSERS + iid[j - 1024]);
    out[i] = acc[node * DIM + d] * 0.25f;
}

// ---------------------------------------------------------------------------
extern "C" void kernel_launch(void* const* d_in, const int* in_sizes, int n_in,
                              void* d_out, int out_size, void* d_ws, size_t ws_size,
                              hipStream_t stream) {
    const float* ue   = (const float*)d_in[0];
    const float* ie   = (const float*)d_in[1];
    const float* vals = (const float*)d_in[2];
    const int*   rows = (const int*)d_in[3];
    const int*   cols = (const int*)d_in[4];
    const int*   uid  = (const int*)d_in[5];
    const int*   iid  = (const int*)d_in[6];

    float* emb  = (float*)d_ws;                               // 38.4 MB
    float* embn = emb  + (size_t)NNODES * DIM;                // 38.4 MB
    float* acc  = embn + (size_t)NNODES * DIM;                // 38.4 MB  (115 MB < 192 MB L2)

    const int init4 = NNODES * (DIM / 4);
    lgcn_init<<<(init4 + 255) / 256, 256, 0, stream>>>(ue, ie, emb, embn, acc);

    const int scatter_blocks = (NEDGES + 7) / 8;              // 8 edge-waves per block
    const int gate_groups    = NNODES / 16;
    const int gate_blocks    = (gate_groups + 7) / 8;

    for (int l = 0; l < NLAYERS; ++l) {
        lgcn_scatter<<<scatter_blocks, 256, 0, stream>>>(vals, rows, cols, emb, embn);
        lgcn_gate<<<gate_blocks, 256, 0, stream>>>(emb, embn, acc);
    }

    lgcn_gather<<<(2048 * DIM + 255) / 256, 256, 0, stream>>>(acc, uid, iid, (float*)d_out);
}